// Phaseformer_1151051235587
// MI455X (gfx1250) — compile-verified
//
#include <hip/hip_runtime.h>
#include <math.h>

typedef __attribute__((ext_vector_type(2))) float v2f;
typedef __attribute__((ext_vector_type(8))) float v8f;

#define DUR 32
#define DIM 256
#define TTOT 528   // 32*33/2
#define LN_EPS 1e-5f

// ---------------------------------------------------------------------------
// Kernel 1: branch convolutions as WMMA f32 GEMMs, K padded per-branch to a
// multiple of 4 so every 16x16x4 K-step sits inside one input channel i.
//
//   Y(256 x 32pad) = W_b(256 x 256*kpad) * Xwin(256*kpad x 32pad)
//   K index kk = i*kpad + t   (A/B zero-masked for t >= k)
//
// grid = 32 branches * 16 row-tiles = 512 blocks, blockDim = 128 (4 waves).
// Wave w handles input channels i in [w*64, w*64+64) (K-split); the four
// partial accumulator tiles are LDS-reduced, then bias + exact GELU + spill
// + branch stats are done cooperatively.
//
// A layout (16x4 f32): lane l holds row M=l%16, K = 2*(l/16)+vgpr  -> the two
//   K values are (t, t+1): one contiguous 8B load from the weight row.
// B layout (4x16 f32): lane l holds col N=l%16, K = 2*(l/16)+vgpr.
// C/D layout (16x16 f32): lane l, vgpr v -> M = v + 8*(l/16), N = l%16.
// ---------------------------------------------------------------------------
__global__ void __launch_bounds__(128) phase_conv_wmma(
    const float* __restrict__ x,      // (1,32,256): x[pos*256 + ch]
    const float* __restrict__ cw,     // (32,256,256,32)
    const float* __restrict__ cb,     // (32,256)
    float* __restrict__ y_ws,         // (32,256,32) padded branch outputs
    float* __restrict__ partials)     // (512,2) per-(b,mtile) {sum, sumsq}
{
  __shared__ float acc[128 * 16];     // 4 waves x 32 lanes x 16 C elements
  __shared__ float wsum[4], wsq[4];

  const int blk   = blockIdx.x;
  const int b     = blk >> 4;                 // branch
  const int mtile = blk & 15;                 // 16-row tile of out channels
  const int k     = b + 1;                    // conv width
  const int kpad  = (k + 3) & ~3;             // K padded per channel
  const int nt    = kpad >> 2;                // 4-wide t-steps per channel
  const int L     = DUR - b;                  // valid output positions

  const int tid  = threadIdx.x;
  const int ks   = tid >> 5;                  // wave id = K-split slice
  const int lane = tid & 31;
  const int half = lane >> 4;                 // K-pair select
  const int lo   = lane & 15;

  const int  orow = mtile * 16 + lo;          // A row (out channel)
  const long wrow = ((long)b * DIM + orow) * (DIM * DUR);

  // per-lane output columns of the two 16-wide N tiles
  const int   p0  = lo, p1 = lo + 16;
  const int   pc0 = (p0 < L) ? p0 : (L - 1);
  const int   pc1 = (p1 < L) ? p1 : (L - 1);
  const float m0  = (p0 < L) ? 1.f : 0.f;
  const float m1  = (p1 < L) ? 1.f : 0.f;

  v8f c0 = {0.f,0.f,0.f,0.f,0.f,0.f,0.f,0.f};
  v8f c1 = {0.f,0.f,0.f,0.f,0.f,0.f,0.f,0.f};

  const int ilo = ks * 64, ihi = ilo + 64;    // this wave's channel slice

  for (int tt = 0; tt < nt; ++tt) {
    // everything t-dependent is hoisted out of the streaming i-loop
    const int   tbase = tt << 2;
    const int   t0 = tbase + 2 * half;
    const int   t1 = t0 + 1;
    const float tm0 = (t0 < k) ? 1.f : 0.f;
    const float tm1 = (t1 < k) ? 1.f : 0.f;
    const int   tc0 = (t0 < k) ? t0 : (k - 1);
    const int   tc1 = (t1 < k) ? t1 : (k - 1);
    const int   xb00 = (pc0 + tc0) * DIM;     // B source rows of x
    const int   xb01 = (pc0 + tc1) * DIM;
    const int   xb10 = (pc1 + tc0) * DIM;
    const int   xb11 = (pc1 + tc1) * DIM;
    const float mb0 = m0, mb1 = m1;
    const long  abase = wrow + tbase + 2 * half;

    for (int i = ilo; i < ihi; ++i) {
      // A: one aligned 8B load (t, t+1 contiguous in the weight row)
      v2f a = *(const v2f*)(cw + abase + (long)i * DUR);
      a.x *= tm0; a.y *= tm1;
      __builtin_prefetch(cw + abase + (long)(i + 64) * DUR, 0, 1);

      v2f b0v, b1v;
      b0v.x = x[xb00 + i] * mb0;
      b0v.y = x[xb01 + i] * mb0;
      b1v.x = x[xb10 + i] * mb1;
      b1v.y = x[xb11 + i] * mb1;

      c0 = __builtin_amdgcn_wmma_f32_16x16x4_f32(false, a, false, b0v,
                                                 (short)0, c0, false, false);
      c1 = __builtin_amdgcn_wmma_f32_16x16x4_f32(false, a, false, b1v,
                                                 (short)0, c1, false, false);
    }
  }

  // spill the 4 partial tiles to LDS (fixed-order reduction across waves)
#pragma unroll
  for (int v = 0; v < 8; ++v) {
    acc[tid * 16 + v]     = c0[v];
    acc[tid * 16 + 8 + v] = c1[v];
  }
  __syncthreads();

  // wave ks post-processes accumulator slots v in [ks*2, ks*2+2) of 8
  float lsum = 0.f, lsq = 0.f;
#pragma unroll
  for (int vo = 0; vo < 2; ++vo) {
    const int v = ks * 2 + vo;
    float y0 = 0.f, y1 = 0.f;
#pragma unroll
    for (int w2 = 0; w2 < 4; ++w2) {
      y0 += acc[(w2 * 32 + lane) * 16 + v];
      y1 += acc[(w2 * 32 + lane) * 16 + 8 + v];
    }
    const int   oo   = mtile * 16 + v + 8 * half;
    const float bias = cb[b * DIM + oo];
    y0 += bias;
    y0 = 0.5f * y0 * (1.0f + erff(y0 * 0.70710678118654752f));
    y1 += bias;
    y1 = 0.5f * y1 * (1.0f + erff(y1 * 0.70710678118654752f));
    const long yb = ((long)(b * DIM + oo)) << 5;   // *32 padded cols
    y_ws[yb + p0] = y0;
    y_ws[yb + p1] = y1;
    lsum += y0 * m0 + y1 * m1;
    lsq  += y0 * y0 * m0 + y1 * y1 * m1;
  }
  // fixed-order wave32 reduction, then combine the 4 waves
  for (int off = 16; off > 0; off >>= 1) {
    lsum += __shfl_down(lsum, off, 32);
    lsq  += __shfl_down(lsq,  off, 32);
  }
  if (lane == 0) { wsum[ks] = lsum; wsq[ks] = lsq; }
  __syncthreads();
  if (tid == 0) {
    partials[2 * blk + 0] = wsum[0] + wsum[1] + wsum[2] + wsum[3];
    partials[2 * blk + 1] = wsq[0] + wsq[1] + wsq[2] + wsq[3];
  }
}

// ---------------------------------------------------------------------------
// Kernel 2: per-branch LN stats + column sums -> c[528].
// grid = 32 (one block per branch), blockDim = 256 (thread = out channel).
// ---------------------------------------------------------------------------
__global__ void __launch_bounds__(256) phase_stats_colsum(
    const float* __restrict__ y_ws,
    const float* __restrict__ partials,
    const float* __restrict__ lnw,    // (32,256,32)
    const float* __restrict__ lnb,    // (32,256,32)
    float* __restrict__ cvec)         // (528,)
{
  __shared__ float red[256];
  __shared__ float s_mu, s_rs;
  const int b   = blockIdx.x;
  const int tid = threadIdx.x;
  const int L   = DUR - b;

  float ps = 0.f, pq = 0.f;
  if (tid < 16) {
    ps = partials[2 * (b * 16 + tid) + 0];
    pq = partials[2 * (b * 16 + tid) + 1];
  }
  red[tid] = ps; __syncthreads();
  for (int s = 128; s > 0; s >>= 1) { if (tid < s) red[tid] += red[tid + s]; __syncthreads(); }
  const float tot = red[0]; __syncthreads();
  red[tid] = pq; __syncthreads();
  for (int s = 128; s > 0; s >>= 1) { if (tid < s) red[tid] += red[tid + s]; __syncthreads(); }
  const float totq = red[0]; __syncthreads();

  if (tid == 0) {
    const float n  = 256.f * (float)L;
    const float mu = tot / n;
    const float vr = totq / n - mu * mu;
    s_mu = mu;
    s_rs = rsqrtf(vr + LN_EPS);
  }
  __syncthreads();
  const float mu = s_mu, rs = s_rs;

  const int off = b * 32 - (b * (b - 1)) / 2;   // branch offset in concat
  const int o   = tid;
  const long yb = ((long)(b * DIM + o)) << 5;
  const long lb = (long)(b * DIM + o) * DUR;
  for (int p = 0; p < L; ++p) {
    const float y = y_ws[yb + p];
    const float v = (y - mu) * rs * lnw[lb + p] + lnb[lb + p];
    red[tid] = v; __syncthreads();
    for (int s = 128; s > 0; s >>= 1) { if (tid < s) red[tid] += red[tid + s]; __syncthreads(); }
    if (tid == 0) cvec[off + p] = red[0];
    __syncthreads();
  }
}

// ---------------------------------------------------------------------------
// Kernel 3: u = c @ Wv^T + 256*bv ; S = u @ out_w^T + 256*out_b ;
//           4 windowed means of S, broadcast to (1,4,256).
// grid = 1, blockDim = 1024.
// ---------------------------------------------------------------------------
__global__ void __launch_bounds__(1024) phase_final(
    const float* __restrict__ cvec,
    const float* __restrict__ ipw,    // (1584,528)
    const float* __restrict__ ipb,    // (1584,)
    const float* __restrict__ opw,    // (528,528)
    const float* __restrict__ opb,    // (528,)
    float* __restrict__ out)          // (1,4,256) = 1024 floats
{
  __shared__ float u[TTOT];
  __shared__ float S[TTOT];
  __shared__ float wm[4];
  const int tid = threadIdx.x;

  if (tid < TTOT) {
    const float* wr = ipw + (long)(2 * TTOT + tid) * TTOT;   // Wv row
    float acc = 256.f * ipb[2 * TTOT + tid];
    for (int kq = 0; kq < TTOT; ++kq) acc += cvec[kq] * wr[kq];
    u[tid] = acc;
  }
  __syncthreads();
  if (tid < TTOT) {
    const float* wr = opw + (long)tid * TTOT;
    float acc = 256.f * opb[tid];
    for (int kq = 0; kq < TTOT; ++kq) acc += u[kq] * wr[kq];
    S[tid] = acc;
  }
  __syncthreads();
  if (tid < 4) {
    float m = 0.f;
    for (int d = 0; d < DUR; ++d) m += S[tid * 4 * DUR + d];  // step = 128
    wm[tid] = m * (1.0f / (float)DUR);
  }
  __syncthreads();
  out[tid] = wm[tid >> 8];   // out[0][w][ch] = wm[w]
}

// ---------------------------------------------------------------------------
extern "C" void kernel_launch(void* const* d_in, const int* in_sizes, int n_in,
                              void* d_out, int out_size, void* d_ws, size_t ws_size,
                              hipStream_t stream) {
  const float* x   = (const float*)d_in[0];
  const float* cw  = (const float*)d_in[1];
  const float* cb  = (const float*)d_in[2];
  const float* lnw = (const float*)d_in[3];
  const float* lnb = (const float*)d_in[4];
  const float* ipw = (const float*)d_in[5];
  const float* ipb = (const float*)d_in[6];
  const float* opw = (const float*)d_in[7];
  const float* opb = (const float*)d_in[8];
  float* out = (float*)d_out;

  float* ws       = (float*)d_ws;
  float* y_ws     = ws;                          // 32*256*32 = 262144 floats
  float* partials = ws + 262144;                 // 512*2     = 1024   floats
  float* cvec     = ws + 262144 + 1024;          // 528 floats

  phase_conv_wmma   <<<dim3(512), dim3(128),  0, stream>>>(x, cw, cb, y_ws, partials);
  phase_stats_colsum<<<dim3(32),  dim3(256),  0, stream>>>(y_ws, partials, lnw, lnb, cvec);
  phase_final       <<<dim3(1),   dim3(1024), 0, stream>>>(cvec, ipw, ipb, opw, opb, out);
}